// CaptionModel_53240414601810
// MI455X (gfx1250) — compile-verified
//
#include <hip/hip_runtime.h>
#include <stdint.h>

// ---------------- Tunables / fixed shapes ----------------
#define WAVE   32
#define NW     16                 // waves per block == K (one wave per beam row)
#define BLOCK  (WAVE * NW)        // 512 threads
#define TOPK   16
#define CHUNK  512                // floats per TDM stream chunk (2 KB)
#define VMAX   50257
#define CNTW   ((VMAX + 3) / 4)   // packed u8 counts, 4 per u32 word

typedef unsigned int       u32;
typedef unsigned long long u64;
typedef unsigned int u32x4 __attribute__((ext_vector_type(4)));
typedef int          i32x8 __attribute__((ext_vector_type(8)));
typedef int          i32x4 __attribute__((ext_vector_type(4)));

static_assert(CHUNK % (4 * WAVE) == 0, "chunk must tile into float4-per-lane passes");

// ---------------- Tensor Data Mover: 1-row (1D-as-2D) tile load into LDS ----------------
// Builds a D# descriptor (groups 0/1; higher groups zero => <=2D tensor) and issues
// TENSOR_LOAD_TO_LDS. Tracked by TENSORcnt; caller waits with s_wait_tensorcnt.
__device__ __forceinline__ void tdm_load_chunk(u32 lds_byte, u64 gaddr, u32 nelem) {
  // Descriptors live in SGPRs: force wave-uniform scalars.
  lds_byte = (u32)__builtin_amdgcn_readfirstlane((int)lds_byte);
  u32 alo  = (u32)__builtin_amdgcn_readfirstlane((int)(u32)(gaddr & 0xffffffffull));
  u32 ahi  = (u32)__builtin_amdgcn_readfirstlane((int)(u32)(gaddr >> 32));
  nelem    = (u32)__builtin_amdgcn_readfirstlane((int)nelem);

  u32x4 g0;
  g0[0] = 1u;                                   // count=1, user mode, no gather
  g0[1] = lds_byte;                             // lds_addr [63:32]
  g0[2] = alo;                                  // global_addr[31:0]
  g0[3] = (ahi & 0x1ffffffu) | (2u << 30);      // global_addr[56:32] | type=2 ("image")

  i32x8 g1;
  g1[0] = (int)(2u << 16);                      // workgroup_mask=0, data_size=2 (4 bytes)
  g1[1] = (int)((nelem & 0xffffu) << 16);       // tensor_dim0[15:0]
  g1[2] = (int)(((nelem >> 16) & 0xffffu) | (1u << 16)); // dim0[31:16] | tensor_dim1=1 (lo)
  g1[3] = (int)((nelem & 0xffffu) << 16);       // tensor_dim1 hi=0 | tile_dim0=nelem
  g1[4] = 1;                                    // tile_dim1=1 | tile_dim2=0
  g1[5] = (int)nelem;                           // tensor_dim0_stride[31:0]
  g1[6] = 0;                                    // stride0[47:32]=0 | stride1 lo=0
  g1[7] = 0;                                    // stride1 hi=0

  i32x4 gz4; gz4[0] = 0; gz4[1] = 0; gz4[2] = 0; gz4[3] = 0;
  i32x8 gz8; gz8[0] = 0; gz8[1] = 0; gz8[2] = 0; gz8[3] = 0;
             gz8[4] = 0; gz8[5] = 0; gz8[6] = 0; gz8[7] = 0;
  // 6-arg toolchain variant: (g0, g1, g2, g3, g4, cpol)
  __builtin_amdgcn_tensor_load_to_lds(g0, g1, gz4, gz4, gz8, 0 /*cpol*/);
}

// ---------------- Kernel ----------------
__global__ __launch_bounds__(BLOCK) void dbs_step_kernel(
    const float* __restrict__ logprobs,        // [B,K,V]
    const int*   __restrict__ beam_seq,        // [B,T,K]
    const float* __restrict__ beam_lps,        // [B,T,K]
    const float* __restrict__ bsum,            // [B,K]
    const float* __restrict__ state,           // [2,2,B,K,H]
    const int*   __restrict__ prev_dec,        // [B,NP,K]
    const int*   __restrict__ t_ptr,           // scalar
    float* __restrict__ out_seq,               // [B,T,K] (float-encoded ints)
    float* __restrict__ out_lps,               // [B,T,K]
    float* __restrict__ out_selp,              // [B,K]
    float* __restrict__ out_state,             // [2,2,B,K,H]
    int B, int K, int V, int T, int H, int NP)
{
  __shared__ u32 s_cnt[CNTW];                          // packed u8 token counts
  __shared__ __align__(16) float s_buf[NW][2][CHUNK];  // per-wave TDM double buffers
  __shared__ float s_topv[NW][TOPK];
  __shared__ int   s_topi[NW][TOPK];
  __shared__ int   s_q[TOPK];
  __shared__ int   s_tok[TOPK];
  __shared__ float s_r[TOPK];
  __shared__ float s_p[TOPK];

  const int b    = blockIdx.x;
  const int tid  = threadIdx.x;
  const int lane = tid & (WAVE - 1);
  const int w    = tid >> 5;          // wave id == beam row (wave32)
  const float NEGINF = -__builtin_inff();

  // ---- Phase 1: diversity counts (packed bytes) in LDS ----
  for (int i = tid; i < CNTW; i += BLOCK) s_cnt[i] = 0u;
  __syncthreads();
  const int ntok = NP * K;            // 256
  for (int i = tid; i < ntok; i += BLOCK) {
    int tok = prev_dec[(size_t)b * ntok + i];
    if (tok >= 0 && tok < V)
      atomicAdd(&s_cnt[(u32)tok >> 2], 1u << ((tok & 3) * 8));
  }
  __syncthreads();

  // ---- Phase 2: per-row streaming top-16 via TDM double buffer ----
  float tv[TOPK]; int ti[TOPK];
  #pragma unroll
  for (int s = 0; s < TOPK; ++s) { tv[s] = NEGINF; ti[s] = 0x7fffffff; }
  float mn = NEGINF; int mnpos = 0;

  const int row = w;
  const u64 rowbase = (u64)(uintptr_t)(logprobs + ((size_t)b * K + row) * V);
  const int nchunks = (V + CHUNK - 1) / CHUNK;
  const u32 lds0 = (u32)(uintptr_t)&s_buf[row][0][0];
  const u32 lds1 = (u32)(uintptr_t)&s_buf[row][1][0];

  {  // prefetch chunk 0
    int len0 = V < CHUNK ? V : CHUNK;
    tdm_load_chunk(lds0, rowbase, (u32)len0);
  }

  auto consider = [&](float val, int v, u32 cntb) {
    if (v < V) {
      float a = val - 0.5f * (float)cntb;      // diversity penalty
      if (v == V - 1) a -= 1000.0f;            // EOS penalty (present in unaug too)
      if (a > mn) {
        #pragma unroll
        for (int s = 0; s < TOPK; ++s) if (s == mnpos) { tv[s] = a; ti[s] = v; }
        mn = tv[0]; mnpos = 0;
        #pragma unroll
        for (int s = 1; s < TOPK; ++s) if (tv[s] < mn) { mn = tv[s]; mnpos = s; }
      }
    }
  };

  for (int c = 0; c < nchunks; ++c) {
    const int vbase = c * CHUNK;
    if (c + 1 < nchunks) {
      int nb = V - (vbase + CHUNK); if (nb > CHUNK) nb = CHUNK;
      tdm_load_chunk((c & 1) ? lds0 : lds1, rowbase + (u64)(vbase + CHUNK) * 4u, (u32)nb);
      __builtin_amdgcn_s_wait_tensorcnt(1);   // chunk c landed (TDM in-order per wave)
    } else {
      __builtin_amdgcn_s_wait_tensorcnt(0);
    }
    asm volatile("" ::: "memory");            // no hoisting of LDS reads above the wait

    const float4* buf4 = (const float4*)&s_buf[row][c & 1][0];
    #pragma unroll
    for (int it = 0; it < CHUNK / (4 * WAVE); ++it) {
      const int e4 = it * WAVE + lane;
      const float4 d = buf4[e4];
      const int v0 = vbase + e4 * 4;          // multiple of 4 -> one count word
      const u32 cw = (v0 < V) ? s_cnt[(u32)v0 >> 2] : 0u;
      consider(d.x, v0 + 0, (cw      ) & 0xffu);
      consider(d.y, v0 + 1, (cw >>  8) & 0xffu);
      consider(d.z, v0 + 2, (cw >> 16) & 0xffu);
      consider(d.w, v0 + 3, (cw >> 24) & 0xffu);
    }
  }

  // ---- Wave-level merge: 32 lanes x 16 -> sorted row top-16 ----
  #pragma unroll 1
  for (int j = 0; j < TOPK; ++j) {
    float bv = NEGINF; int bi = 0x7fffffff; int bs = -1;
    #pragma unroll
    for (int s = 0; s < TOPK; ++s) {
      bool better = (tv[s] > bv) || (tv[s] == bv && ti[s] < bi);
      if (better) { bv = tv[s]; bi = ti[s]; bs = s; }
    }
    #pragma unroll
    for (int off = 16; off >= 1; off >>= 1) {
      float ov = __shfl_xor(bv, off, WAVE);
      int   oi = __shfl_xor(bi, off, WAVE);
      if (ov > bv || (ov == bv && oi < bi)) { bv = ov; bi = oi; bs = -1; }
    }
    if (bs >= 0) {  // unique winner (tokens unique per row): clear my slot
      #pragma unroll
      for (int s = 0; s < TOPK; ++s) if (s == bs) { tv[s] = NEGINF; ti[s] = 0x7fffffff; }
    }
    if (lane == 0) { s_topv[row][j] = bv; s_topi[row][j] = bi; }
  }
  __syncthreads();

  // ---- Phase 3: global top-16 over K x K candidates (wave 0) ----
  if (w == 0) {
    float cv[8]; int cf[8];
    const int k0 = lane >> 1;                     // 8 consecutive flats share one beam
    const float bs0 = bsum[(size_t)b * K + k0];
    #pragma unroll
    for (int s = 0; s < 8; ++s) {
      int flat = lane * 8 + s;                    // flat = k*16 + j
      cv[s] = bs0 + s_topv[k0][flat & 15];
      cf[s] = flat;
    }
    #pragma unroll 1
    for (int j = 0; j < TOPK; ++j) {
      float bv = NEGINF; int bf = 0x7fffffff; int bsl = -1;
      #pragma unroll
      for (int s = 0; s < 8; ++s) {
        bool better = (cv[s] > bv) || (cv[s] == bv && cf[s] < bf);
        if (better) { bv = cv[s]; bf = cf[s]; bsl = s; }
      }
      #pragma unroll
      for (int off = 16; off >= 1; off >>= 1) {
        float ov = __shfl_xor(bv, off, WAVE);
        int   of = __shfl_xor(bf, off, WAVE);
        if (ov > bv || (ov == bv && of < bf)) { bv = ov; bf = of; bsl = -1; }
      }
      if (bsl >= 0) {
        #pragma unroll
        for (int s = 0; s < 8; ++s) if (s == bsl) { cv[s] = NEGINF; cf[s] = 0x7fffffff; }
      }
      if (lane == 0) {
        int k = bf >> 4, jj = bf & 15;
        int tok = s_topi[k][jj];
        u32 cwd = (tok >= 0 && tok < V) ? s_cnt[(u32)tok >> 2] : 0u;
        float cnt = (float)((cwd >> ((tok & 3) * 8)) & 0xffu);
        s_q[j]   = k;
        s_tok[j] = tok;
        s_p[j]   = bv;
        s_r[j]   = s_topv[k][jj] + 0.5f * cnt;   // unaug = aug + lambda*count (EOS kept)
      }
    }
  }
  __syncthreads();

  // ---- Phase 4: write outputs ----
  const int t = *t_ptr;
  for (int i = tid; i < T * K; i += BLOCK) {
    const int r0 = i / K, j = i % K;
    const size_t base = ((size_t)b * T + r0) * K;
    int tokv; float lv;
    if (r0 < t)       { int q = s_q[j]; tokv = beam_seq[base + q]; lv = beam_lps[base + q]; }
    else if (r0 == t) { tokv = s_tok[j];                           lv = s_r[j]; }
    else              { tokv = beam_seq[base + j];                 lv = beam_lps[base + j]; }
    out_seq[base + j] = (float)tokv;
    out_lps[base + j] = lv;
  }
  if (tid < K) out_selp[(size_t)b * K + tid] = s_p[tid];

  // state gather along beam dim: out[s,l,b,j,:] = state[s,l,b,q[j],:]  (float4 copies)
  const int H4 = H >> 2;
  const int n4 = 4 * K * H4;                      // (S*L)=4 slices per batch
  const float4* st4 = (const float4*)state;
  float4* os4 = (float4*)out_state;
  for (int i = tid; i < n4; i += BLOCK) {
    const int h4 = i % H4;
    const int rest = i / H4;
    const int k  = rest % K;
    const int sl = rest / K;                      // 0..3 == (s*2+l)
    const int q  = s_q[k];
    const size_t src = (((size_t)sl * B + b) * K + q) * H4 + h4;
    const size_t dst = (((size_t)sl * B + b) * K + k) * H4 + h4;
    os4[dst] = st4[src];
  }
}

// ---------------- Host launcher ----------------
extern "C" void kernel_launch(void* const* d_in, const int* in_sizes, int n_in,
                              void* d_out, int out_size, void* d_ws, size_t ws_size,
                              hipStream_t stream) {
  const float* logprobs = (const float*)d_in[0];
  const int*   beam_seq = (const int*)  d_in[1];
  const float* beam_lps = (const float*)d_in[2];
  const float* bsum     = (const float*)d_in[3];
  const float* state    = (const float*)d_in[4];
  const int*   prev_dec = (const int*)  d_in[5];
  const int*   t_ptr    = (const int*)  d_in[6];

  const int B = 128, K = 16;
  const int V  = in_sizes[0] / (B * K);       // 50257
  const int T  = in_sizes[1] / (B * K);       // 20
  const int H  = in_sizes[4] / (4 * B * K);   // 512
  const int NP = in_sizes[5] / (B * K);       // 16

  float* out       = (float*)d_out;
  float* out_seq   = out;
  float* out_lps   = out_seq  + (size_t)B * T * K;
  float* out_selp  = out_lps  + (size_t)B * T * K;
  float* out_state = out_selp + (size_t)B * K;

  dbs_step_kernel<<<B, BLOCK, 0, stream>>>(
      logprobs, beam_seq, beam_lps, bsum, state, prev_dec, t_ptr,
      out_seq, out_lps, out_selp, out_state, B, K, V, T, H, NP);
}